// MyGaussianSimpleFast_1623497637993
// MI455X (gfx1250) — compile-verified
//
#include <hip/hip_runtime.h>
#include <hip/hip_bf16.h>
#include <math.h>

typedef __attribute__((ext_vector_type(2))) float v2f;
typedef __attribute__((ext_vector_type(8))) float v8f;

#define WIDTH 96
#define HEIGHT 96
#define NPIX (WIDTH * HEIGHT)      // 9216 pixels == 9216 gaussians
#define FEATC 64
#define GDIM 8
#define TW 6
#define TH 6
#define LN255 5.5412635f           // ln(255): alpha >= 1/255  <=>  sigma <= ln(255)

__device__ __forceinline__ float sigmoidf_(float x) {
    return 1.0f / (1.0f + __expf(-x));
}

// ---------------------------------------------------------------------------
// Kernel 1: fused conv3x3(3->64) + head(64->8) + gaussian parameterization.
// Emits per gaussian n:
//   gfeat[n][0..7]: rank-6 bilinear features (padded to 8)
//   grgb[n] = {r, g, b, packed tile bounds (tminx|tmaxx<<8|tminy<<16|tmaxy<<24)}
// ---------------------------------------------------------------------------
__global__ __launch_bounds__(256) void gauss_param_kernel(
    const float* __restrict__ inp, const float* __restrict__ w_enc,
    const float* __restrict__ b_enc, const float* __restrict__ w_head,
    const float* __restrict__ b_head, float* __restrict__ gfeat,
    float* __restrict__ grgb) {
  __shared__ float s_wenc[FEATC * 27];
  __shared__ float s_benc[FEATC];
  __shared__ float s_whead[GDIM * FEATC];
  __shared__ float s_bhead[GDIM];
  for (int i = threadIdx.x; i < FEATC * 27; i += 256) s_wenc[i] = w_enc[i];
  for (int i = threadIdx.x; i < FEATC; i += 256) s_benc[i] = b_enc[i];
  for (int i = threadIdx.x; i < GDIM * FEATC; i += 256) s_whead[i] = w_head[i];
  if (threadIdx.x < GDIM) s_bhead[threadIdx.x] = b_head[threadIdx.x];
  __syncthreads();

  const int n = blockIdx.x * 256 + threadIdx.x;   // grid sized exactly: no guard
  const int x = n % WIDTH;
  const int y = n / WIDTH;

  // 3x3x3 input patch, zero-padded (SAME)
  float patch[27];
#pragma unroll
  for (int ic = 0; ic < 3; ++ic)
#pragma unroll
    for (int dy = 0; dy < 3; ++dy)
#pragma unroll
      for (int dx = 0; dx < 3; ++dx) {
        const int yy = y + dy - 1, xx = x + dx - 1;
        float v = 0.0f;
        if (yy >= 0 && yy < HEIGHT && xx >= 0 && xx < WIDTH)
          v = inp[ic * NPIX + yy * WIDTH + xx];
        patch[ic * 9 + dy * 3 + dx] = v;
      }

  float pred[GDIM];
#pragma unroll
  for (int o = 0; o < GDIM; ++o) pred[o] = s_bhead[o];
  for (int oc = 0; oc < FEATC; ++oc) {
    float f = s_benc[oc];
#pragma unroll
    for (int k = 0; k < 27; ++k) f = fmaf(patch[k], s_wenc[oc * 27 + k], f);
#pragma unroll
    for (int o = 0; o < GDIM; ++o) pred[o] = fmaf(s_whead[o * FEATC + oc], f, pred[o]);
  }

  // gaussian parameterization
  const float theta = sigmoidf_(pred[3]) * 6.2831853071795864f;
  const float sc0 = sigmoidf_(pred[4]) * 0.5f + 1e-6f;
  const float sc1 = sigmoidf_(pred[5]) * 0.5f + 1e-6f;
  const float offx = tanhf(pred[6]);
  const float offy = tanhf(pred[7]);
  float sth, cth;
  __sincosf(theta, &sth, &cth);
  const float v0 = sc0 * sc0, v1 = sc1 * sc1;
  const float S00 = cth * cth * v0 + sth * sth * v1;
  const float S01 = cth * sth * (v0 - v1);
  const float S11 = sth * sth * v0 + cth * cth * v1;
  const float det = S00 * S11 - S01 * S01;
  const float inv = 1.0f / det;
  const float ca = S11 * inv, cb = -S01 * inv, cc = S00 * inv;
  const float bmid = 0.5f * (S00 + S11);
  const float lam1 = bmid + sqrtf(fmaxf(0.1f, bmid * bmid - det));
  const float radius = ceilf(3.0f * sqrtf(lam1));

  const float x_ndc = (2.0f * (float)x / 96.0f - 1.0f) + 2.0f * offx / 96.0f - 1.0f / 96.0f;
  const float y_ndc = (2.0f * (float)y / 96.0f - 1.0f) + 2.0f * offy / 96.0f - 1.0f / 96.0f;
  const float cx = 0.5f * 96.0f * (x_ndc + 1.0f) - 0.5f;
  const float cy = 0.5f * 96.0f * (y_ndc + 1.0f) - 0.5f;

  int tminx, tmaxx, tminy, tmaxy;
  if (radius > 0.0f) {
    tminx = min(max((int)((cx - radius) / 16.0f), 0), TW);
    tmaxx = min(max((int)((cx + radius) / 16.0f) + 1, 0), TW);
    tminy = min(max((int)((cy - radius) / 16.0f), 0), TH);
    tmaxy = min(max((int)((cy + radius) / 16.0f) + 1, 0), TH);
  } else {  // radius<=0: empty range -> all pairs invalid
    tminx = 255; tmaxx = 0; tminy = 255; tmaxy = 0;
  }
  const unsigned bw = (unsigned)tminx | ((unsigned)tmaxx << 8) |
                      ((unsigned)tminy << 16) | ((unsigned)tmaxy << 24);

  // rank-6 features: sigma[p,g] = f0*px^2 + f1*px + f2*py^2 + f3*py + f4*px*py + f5
  float* gf = gfeat + (size_t)n * 8;
  gf[0] = 0.5f * ca;
  gf[1] = -(ca * cx + cb * cy);
  gf[2] = 0.5f * cc;
  gf[3] = -(cc * cy + cb * cx);
  gf[4] = cb;
  gf[5] = 0.5f * ca * cx * cx + 0.5f * cc * cy * cy + cb * cx * cy;
  gf[6] = 0.0f;
  gf[7] = 0.0f;

  float4 r;
  r.x = pred[0]; r.y = pred[1]; r.z = pred[2];
  r.w = __uint_as_float(bw);
  *(float4*)(grgb + (size_t)n * 4) = r;
}

// ---------------------------------------------------------------------------
// Kernel 2: rasterize via f32 WMMA.
// Wave owns 16 consecutive pixels (one M-tile, constant A fragment).
// Loop over gaussians in tiles of 16 (N dim): sigma tile = two
// v_wmma_f32_16x16x4_f32, then elementwise exp/mask/rgb-accumulate.
// ---------------------------------------------------------------------------
__global__ __launch_bounds__(256) void gauss_raster_kernel(
    const float* __restrict__ gfeat, const float* __restrict__ grgb,
    float* __restrict__ out) {
  const int lane = threadIdx.x & 31;
  const int wave_id = blockIdx.x * 8 + (threadIdx.x >> 5);
  const int tile_base = wave_id * 16;      // 16 consecutive pixels, same image row
  const bool hi = lane >= 16;

  // A fragment (pixel features), loop-invariant.
  // Lane L: M = L%16, K pair = (L/16)*2.  Features: [px^2,px,py^2,py,px*py,1,0,0]
  {
  }
  const int m = lane & 15;
  const int p0 = tile_base + m;
  const float px = (float)(p0 % WIDTH);
  const float py = (float)(p0 / WIDTH);
  v2f a1, a2;
  if (!hi) { a1.x = px * px; a1.y = px;  a2.x = px * py; a2.y = 1.0f; }
  else     { a1.x = py * py; a1.y = py;  a2.x = 0.0f;    a2.y = 0.0f; }

  // tile coords: uniform over the wave (96 % 16 == 0)
  const int tx = (tile_base % WIDTH) >> 4;
  const int ty = (tile_base / WIDTH) >> 4;
  const int kb = hi ? 2 : 0;

  float acc[8][3];
#pragma unroll
  for (int v = 0; v < 8; ++v) { acc[v][0] = 0.f; acc[v][1] = 0.f; acc[v][2] = 0.f; }

  for (int t = 0; t < NPIX; t += 16) {
    const int g = t + (lane & 15);
    const float* fp = gfeat + (size_t)g * 8;
    const v2f b1 = *(const v2f*)(fp + kb);       // K = kb, kb+1
    const v2f b2 = *(const v2f*)(fp + kb + 4);   // K = kb+4, kb+5
    const float4 rb = *(const float4*)(grgb + (size_t)g * 4);
    const unsigned bw = __float_as_uint(rb.w);
    const int tminx = bw & 0xFF, tmaxx = (bw >> 8) & 0xFF;
    const int tminy = (bw >> 16) & 0xFF, tmaxy = (bw >> 24) & 0xFF;
    const bool bok = (tx >= tminx) & (tx < tmaxx) & (ty >= tminy) & (ty < tmaxy);

    v8f c = {};
    c = __builtin_amdgcn_wmma_f32_16x16x4_f32(false, a1, false, b1, (short)0, c,
                                              false, false);
    c = __builtin_amdgcn_wmma_f32_16x16x4_f32(false, a2, false, b2, (short)0, c,
                                              false, false);
#pragma unroll
    for (int v = 0; v < 8; ++v) {
      const float sigma = c[v];
      float w = fminf(0.999f, __expf(-sigma));
      const bool ok = bok && (sigma >= 0.0f) && (sigma <= LN255);
      w = ok ? w : 0.0f;
      acc[v][0] = fmaf(w, rb.x, acc[v][0]);
      acc[v][1] = fmaf(w, rb.y, acc[v][1]);
      acc[v][2] = fmaf(w, rb.z, acc[v][2]);
    }
    if (t + 256 < NPIX)
      __builtin_prefetch(gfeat + (size_t)(t + 256) * 8, 0, 1);  // global_prefetch_b8
  }

  // Reduce across the 16 gaussian-column lanes within each half-wave.
#pragma unroll
  for (int v = 0; v < 8; ++v)
#pragma unroll
    for (int ch = 0; ch < 3; ++ch) {
      float s = acc[v][ch];
      s += __shfl_xor(s, 1, 32);
      s += __shfl_xor(s, 2, 32);
      s += __shfl_xor(s, 4, 32);
      s += __shfl_xor(s, 8, 32);
      acc[v][ch] = s;
    }

  if ((lane & 15) == 0) {
    const int prow = tile_base + (hi ? 8 : 0);   // lane 0 -> pixels 0..7, lane 16 -> 8..15
#pragma unroll
    for (int v = 0; v < 8; ++v) {
      const int pp = prow + v;
#pragma unroll
      for (int ch = 0; ch < 3; ++ch)
        out[ch * NPIX + pp] = fminf(1.0f, fmaxf(0.0f, acc[v][ch]));
    }
  }
}

// ---------------------------------------------------------------------------
extern "C" void kernel_launch(void* const* d_in, const int* in_sizes, int n_in,
                              void* d_out, int out_size, void* d_ws, size_t ws_size,
                              hipStream_t stream) {
  const float* inp    = (const float*)d_in[0];
  const float* w_enc  = (const float*)d_in[1];
  const float* b_enc  = (const float*)d_in[2];
  const float* w_head = (const float*)d_in[3];
  const float* b_head = (const float*)d_in[4];
  float* out = (float*)d_out;

  float* gfeat = (float*)d_ws;                 // NPIX * 8 floats
  float* grgb  = gfeat + (size_t)NPIX * 8;     // NPIX * 4 floats

  // 9216 gaussians / 256 threads = 36 blocks
  gauss_param_kernel<<<36, 256, 0, stream>>>(inp, w_enc, b_enc, w_head, b_head,
                                             gfeat, grgb);
  // 9216 pixels / (8 waves * 16 px) = 72 blocks
  gauss_raster_kernel<<<72, 256, 0, stream>>>(gfeat, grgb, out);
}